// SOM_49349174231446
// MI455X (gfx1250) — compile-verified
//
#include <hip/hip_runtime.h>

// SOM training step for MI455X (gfx1250), wave32 + V_WMMA_F32_16X16X4_F32.
//
// Shapes: X (2048 x 256) f32, W (1024 x 256) f32, loc (1024 x 2) f32,
//         passo/total scalar ints. Output: new W (1024 x 256) f32.
//
// Roofline: ~2.2 GFLOP fp32, ~4 MB HBM traffic -> fully L2-resident (192 MB),
// compute/latency bound. fp32 WMMA (16x16x4) keeps the BMU argmin exact
// (bf16/f16 WMMA would risk flipping argmin winners).
//
// Both GEMMs use a software-pipelined K loop: 32-deep K chunks are loaded
// into a shadow register buffer (16x global_load_b64) while the WMMA chain
// (8x v_wmma back-to-back, accumulator-chained) consumes the other buffer.

typedef __attribute__((ext_vector_type(2))) float v2f;
typedef __attribute__((ext_vector_type(8))) float v8f;

#define NEURONS 1024
#define DIM     256
#define BATCH   2048
#define SIGMA0  16.0f
#define TAXA0   0.5f

// ---------------------------------------------------------------- w2[n] = sum_d W[n,d]^2
__global__ __launch_bounds__(256) void som_w2(const float* __restrict__ W,
                                              float* __restrict__ w2) {
    int n = blockIdx.x * blockDim.x + threadIdx.x;   // 1024 threads total
    const float* row = W + n * DIM;
    float s = 0.0f;
#pragma unroll 8
    for (int d = 0; d < DIM; ++d) s += row[d] * row[d];
    w2[n] = s;
}

// ---------------------------------------------------------------- Xt[d, b] = X[b, d]
// Column-major copy of X so the update GEMM's B-operand is a contiguous
// float2 per lane (matches the 4x16 B layout: row K striped, col = lane%16).
__global__ __launch_bounds__(256) void som_transpose_x(const float* __restrict__ X,
                                                       float* __restrict__ Xt) {
    int tid = blockIdx.x * blockDim.x + threadIdx.x;   // 512K threads
    int d = tid >> 11;          // / BATCH
    int b = tid & (BATCH - 1);
    Xt[(size_t)d * BATCH + b] = X[(size_t)b * DIM + d];
}

// ------------------------------------- S = X @ W^T via WMMA f32 16x16x4, fused argmin
// One block = 16 batch rows; 8 waves each cover 8 of the 64 neuron tiles.
// A-lane (16x4 f32): row M = lane%16, k = 2*(lane/16)+{0,1}  -> float2 from X.
// B-lane (4x16 f32): row K = 2*(lane/16)+{0,1}, col N = lane%16; B = W^T so this
//                    is a float2 from W's row (nbase+lane%16).
// C/D layout: VGPR r holds (row = r + 8*(lane/16), col = lane%16).
__global__ __launch_bounds__(256) void som_scores_argmin(const float* __restrict__ X,
                                                         const float* __restrict__ W,
                                                         const float* __restrict__ w2,
                                                         int* __restrict__ bmu) {
    __shared__ float s_val[8][16];
    __shared__ int   s_idx[8][16];

    const int lane  = threadIdx.x & 31;
    const int wave  = threadIdx.x >> 5;
    const int l16   = lane & 15;
    const int half  = lane >> 4;
    const int bbase = blockIdx.x * 16;

    const float* Arow = X + (size_t)(bbase + l16) * DIM + 2 * half;

    float bestv[8];
    int   besti[8];
#pragma unroll
    for (int r = 0; r < 8; ++r) { bestv[r] = 3.4e38f; besti[r] = 0; }

    for (int t = 0; t < 8; ++t) {
        const int nbase = (wave + 8 * t) * 16;
        const float* Brow = W + (size_t)(nbase + l16) * DIM + 2 * half;

        // warm L2/L0 for the next neuron tile of this wave (speculative,
        // dropped if out of range) -> global_prefetch_b8
        __builtin_prefetch(Brow + 128 * DIM, 0, 0);

        v8f acc = {};

        auto loadc = [&](v2f* a, v2f* b, int c) {
            const float* ap = Arow + c * 32;
            const float* bp = Brow + c * 32;
#pragma unroll
            for (int u = 0; u < 8; ++u) {
                a[u] = *(const v2f*)(ap + 4 * u);
                b[u] = *(const v2f*)(bp + 4 * u);
            }
        };
        auto mm = [&](v2f* a, v2f* b) {
#pragma unroll
            for (int u = 0; u < 8; ++u)
                acc = __builtin_amdgcn_wmma_f32_16x16x4_f32(
                    false, a[u], false, b[u], (short)0, acc, false, false);
        };

        // software pipeline: 8 chunks of K=32, double-buffered
        v2f a0[8], b0[8], a1[8], b1[8];
        loadc(a0, b0, 0);
        for (int c = 0; c < 6; c += 2) {
            loadc(a1, b1, c + 1);
            mm(a0, b0);
            loadc(a0, b0, c + 2);
            mm(a1, b1);
        }
        loadc(a1, b1, 7);
        mm(a0, b0);
        mm(a1, b1);

        // score(b, n) relative argmin metric: w2[n] - 2 * (X . W[n])
        const float w2l  = w2[nbase + l16];
        const int   nidx = nbase + l16;
#pragma unroll
        for (int r = 0; r < 8; ++r) {
            float v = w2l - 2.0f * acc[r];
            if (v < bestv[r]) { bestv[r] = v; besti[r] = nidx; }
        }
    }

    // reduce across the 16 lanes of each half (rows 0-7 live in lanes 0-15,
    // rows 8-15 in lanes 16-31); xor masks 1..8 stay within each half.
#pragma unroll
    for (int m = 8; m >= 1; m >>= 1) {
#pragma unroll
        for (int r = 0; r < 8; ++r) {
            float ov = __shfl_xor(bestv[r], m, 32);
            int   oi = __shfl_xor(besti[r], m, 32);
            if (ov < bestv[r] || (ov == bestv[r] && oi < besti[r])) {
                bestv[r] = ov; besti[r] = oi;
            }
        }
    }

    if (l16 == 0) {
#pragma unroll
        for (int r = 0; r < 8; ++r) {
            s_val[wave][r + 8 * half] = bestv[r];
            s_idx[wave][r + 8 * half] = besti[r];
        }
    }
    __syncthreads();

    if (threadIdx.x < 16) {
        float bv = s_val[0][threadIdx.x];
        int   bi = s_idx[0][threadIdx.x];
#pragma unroll
        for (int wv = 1; wv < 8; ++wv) {
            float v = s_val[wv][threadIdx.x];
            int   i = s_idx[wv][threadIdx.x];
            if (v < bv || (v == bv && i < bi)) { bv = v; bi = i; }
        }
        bmu[bbase + threadIdx.x] = bi;
    }
}

// ------------------------------------------------- Ht[n, b] = exp(-||loc n - loc bmu||^2 / 2s^2)
// Stored N-major so the update GEMM's A-matrix loads are contiguous per lane.
__global__ __launch_bounds__(256) void som_build_h(const float* __restrict__ loc,
                                                   const int* __restrict__ bmu,
                                                   const int* __restrict__ passo,
                                                   const int* __restrict__ total,
                                                   float* __restrict__ Ht) {
    int tid = blockIdx.x * blockDim.x + threadIdx.x;
    int n = tid >> 11;          // / BATCH
    int b = tid & (BATCH - 1);

    float frac   = (float)passo[0] / (float)total[0];
    float sigma  = SIGMA0 * expf(-frac);
    float inv2s2 = 1.0f / (2.0f * sigma * sigma);

    int   m  = bmu[b];
    float dx = loc[2 * n]     - loc[2 * m];
    float dy = loc[2 * n + 1] - loc[2 * m + 1];
    Ht[(size_t)n * BATCH + b] = expf(-(dx * dx + dy * dy) * inv2s2);
}

// ------------------------------------- out = W + taxa/B * (H^T @ X - rowsum(H) * W)
// One wave per 16x16 output tile (64 n-tiles x 16 d-tiles = 1024 waves = 128 blocks).
// A = H^T from Ht (N-major, contiguous float2), B = X from Xt (D-major,
// contiguous float2). rowsum(H) comes for free: accumulating a.x+a.y over the
// K loop gives, for lane l, the partial sum of row (lane%16) over its k-half;
// lanes l and l+16 cover disjoint halves, so one xor-16 add completes it.
__global__ __launch_bounds__(256) void som_update(const float* __restrict__ Ht,
                                                  const float* __restrict__ Xt,
                                                  const float* __restrict__ W,
                                                  const int* __restrict__ passo,
                                                  const int* __restrict__ total,
                                                  float* __restrict__ out) {
    const int lane  = threadIdx.x & 31;
    const int wave  = threadIdx.x >> 5;
    const int l16   = lane & 15;
    const int half  = lane >> 4;
    const int gw    = blockIdx.x * 8 + wave;   // 0..1023
    const int nbase = (gw >> 4) * 16;          // 64 neuron tiles
    const int dbase = (gw & 15) * 16;          // 16 dim tiles

    const float* Arow = Ht + (size_t)(nbase + l16) * BATCH + 2 * half;
    const float* Brow = Xt + (size_t)(dbase + l16) * BATCH + 2 * half;

    v8f   acc  = {};
    float hacc = 0.0f;

    auto loadc = [&](v2f* a, v2f* b, int c) {
        const float* ap = Arow + c * 32;
        const float* bp = Brow + c * 32;
#pragma unroll
        for (int u = 0; u < 8; ++u) {
            a[u] = *(const v2f*)(ap + 4 * u);
            b[u] = *(const v2f*)(bp + 4 * u);
        }
    };
    auto mm = [&](v2f* a, v2f* b) {
#pragma unroll
        for (int u = 0; u < 8; ++u) {
            hacc += a[u].x + a[u].y;
            acc = __builtin_amdgcn_wmma_f32_16x16x4_f32(
                false, a[u], false, b[u], (short)0, acc, false, false);
        }
    };

    // software pipeline over K=2048: 64 chunks of 32, double-buffered
    v2f a0[8], b0[8], a1[8], b1[8];
    loadc(a0, b0, 0);
    for (int c = 0; c < 62; c += 2) {
        loadc(a1, b1, c + 1);
        mm(a0, b0);
        loadc(a0, b0, c + 2);
        mm(a1, b1);
    }
    loadc(a1, b1, 63);
    mm(a0, b0);
    mm(a1, b1);

    // complete rowsum(H): lane i and i+16 hold the two k-halves of row i%16
    hacc += __shfl_xor(hacc, 16, 32);

    float frac  = (float)passo[0] / (float)total[0];
    float taxa  = TAXA0 * expf(-frac);
    float scale = taxa / (float)BATCH;

#pragma unroll
    for (int r = 0; r < 8; ++r) {
        int   m  = nbase + r + 8 * half;           // output row for VGPR r
        float hs = __shfl(hacc, r + 8 * half, 32); // hsum[m - nbase]
        size_t o = (size_t)m * DIM + dbase + l16;
        float  w = W[o];
        out[o] = w + scale * (acc[r] - hs * w);
    }
}

extern "C" void kernel_launch(void* const* d_in, const int* in_sizes, int n_in,
                              void* d_out, int out_size, void* d_ws, size_t ws_size,
                              hipStream_t stream) {
    const float* X     = (const float*)d_in[0];
    const float* W     = (const float*)d_in[1];
    const float* loc   = (const float*)d_in[2];
    const int*   passo = (const int*)d_in[3];
    const int*   total = (const int*)d_in[4];
    float*       out   = (float*)d_out;

    // workspace layout: Ht (8 MB) | Xt (2 MB) | bmu (8 KB) | w2 (4 KB)
    char*  ws  = (char*)d_ws;
    float* Ht  = (float*)ws;
    float* Xt  = (float*)(ws + (size_t)NEURONS * BATCH * sizeof(float));
    int*   bmu = (int*)(ws + (size_t)(NEURONS + DIM) * BATCH * sizeof(float));
    float* w2  = (float*)(ws + (size_t)(NEURONS + DIM) * BATCH * sizeof(float)
                             + (size_t)BATCH * sizeof(int));

    som_w2<<<NEURONS / 256, 256, 0, stream>>>(W, w2);
    som_transpose_x<<<(DIM * BATCH) / 256, 256, 0, stream>>>(X, Xt);
    som_scores_argmin<<<BATCH / 16, 256, 0, stream>>>(X, W, w2, bmu);
    som_build_h<<<(NEURONS * BATCH) / 256, 256, 0, stream>>>(loc, bmu, passo, total, Ht);
    som_update<<<(NEURONS * 16) / 128, 256, 0, stream>>>(Ht, Xt, W, passo, total, out);
}